// VGAEModel_CL_58961311039544
// MI455X (gfx1250) — compile-verified
//
#include <hip/hip_runtime.h>

typedef float v2f __attribute__((ext_vector_type(2)));
typedef float v8f __attribute__((ext_vector_type(8)));

#define N_NODES 65536
#define N_EDGES 2097152

// ---------------------------------------------------------------- utilities

__global__ void zero_f32(float* __restrict__ p, size_t n) {
    size_t i = (size_t)blockIdx.x * blockDim.x + threadIdx.x;
    size_t stride = (size_t)gridDim.x * blockDim.x;
    for (; i < n; i += stride) p[i] = 0.0f;
}

__global__ void degree_kernel(const int* __restrict__ src, const int* __restrict__ dst,
                              float* deg_out, float* deg_in, int nE) {
    int i = blockIdx.x * blockDim.x + threadIdx.x;
    if (i < nE) {
        atomicAdd(deg_out + src[i], 1.0f);
        atomicAdd(deg_in  + dst[i], 1.0f);
    }
}

__global__ void invsqrt_kernel(float* c, int n) {
    int i = blockIdx.x * blockDim.x + threadIdx.x;
    if (i < n) c[i] = rsqrtf(fmaxf(c[i], 1.0f));
}

// ------------------------------------------------- fp32 WMMA GEMM (16x16x4)
//
// C[M, NT*16] = rowscale[m] * (A[M,K] @ W[K, NT*16])
// One wave per 16-row strip; the wave keeps NT 16x16 f32 accumulators (v8f
// each) and reuses a single A fragment across all NT WMMAs per K-step.
//
// Fragment layouts per CDNA5 ISA 7.12.2:
//   A (16x4 f32, 2 VGPRs): lanes 0-15 row M=lane, K={0,1}; lanes 16-31 K={2,3}
//   B ( 4x16 f32, 2 VGPRs): lanes 0-15 col N=lane, K rows {0,1}; lanes 16-31 {2,3}
//   C/D (16x16 f32, 8 VGPRs): VGPR r -> M = r + 8*(lane>=16), N = lane&15
template <int K, int NT>
__global__ __launch_bounds__(256) void gemm_f32_wmma(
        const float* __restrict__ A, const float* __restrict__ W,
        const float* __restrict__ rowscale, float* __restrict__ C) {
    const int NO   = NT * 16;
    const int lane = threadIdx.x & 31;
    const int wave = threadIdx.x >> 5;
    const int half = lane >> 4;   // 0: lanes 0-15, 1: lanes 16-31
    const int l16  = lane & 15;
    const int mbase = (blockIdx.x * 8 + wave) * 16;

    const float* arow = A + (size_t)(mbase + l16) * K;

    v8f acc[NT] = {};

    for (int k = 0; k < K; k += 4) {
        // A fragment: rows of this wave's strip, K columns {k+2*half, k+2*half+1}
        v2f a = *(const v2f*)(arow + k + half * 2);
#pragma unroll
        for (int j = 0; j < NT; ++j) {
            const int ncol = j * 16 + l16;
            v2f b;
            b.x = W[(size_t)(k + half * 2 + 0) * NO + ncol];
            b.y = W[(size_t)(k + half * 2 + 1) * NO + ncol];
            acc[j] = __builtin_amdgcn_wmma_f32_16x16x4_f32(
                false, a, false, b, (short)0, acc[j], false, false);
        }
    }

#pragma unroll
    for (int j = 0; j < NT; ++j) {
        const int ncol = j * 16 + l16;
#pragma unroll
        for (int r = 0; r < 8; ++r) {
            const int row = mbase + half * 8 + r;
            C[(size_t)row * NO + ncol] = acc[j][r] * rowscale[row];
        }
    }
}

// -------------------------------------------------- edge scatter (segment_sum)
// width = W4*4 floats per node row; W4 lanes per edge, one float4 per lane.
// Aggregation buffer is L2-resident (<=32MB), atomics resolve in L2.
template <int W4>
__global__ void scatter_add(const float* __restrict__ msg,
                            const int* __restrict__ src, const int* __restrict__ dst,
                            float* agg, int nE) {
    const int WID = W4 * 4;
    int t = blockIdx.x * blockDim.x + threadIdx.x;
    int e = t / W4;
    int c = (t % W4) * 4;
    if (e < nE) {
        int s = src[e], d = dst[e];
        float4 v = *(const float4*)(msg + (size_t)s * WID + c);
        float* p = agg + (size_t)d * WID + c;
        atomicAdd(p + 0, v.x);
        atomicAdd(p + 1, v.y);
        atomicAdd(p + 2, v.z);
        atomicAdd(p + 3, v.w);
    }
}

// ------------------------------------------------------------ pointwise fusions

template <int WIDTH>
__global__ void bias_relu_inplace(float* h, const float* __restrict__ c_in,
                                  const float* __restrict__ b, size_t total) {
    size_t i = (size_t)blockIdx.x * blockDim.x + threadIdx.x;
    if (i < total) {
        int row = (int)(i / WIDTH);
        int col = (int)(i % WIDTH);
        float v = h[i] * c_in[row] + b[col];
        h[i] = fmaxf(v, 0.0f);
    }
}

// z = (agg2*c_in + b2) + noise * exp(agg3*c_in + b3)
__global__ void reparam_kernel(const float* __restrict__ agg2, const float* __restrict__ agg3,
                               const float* __restrict__ c_in,
                               const float* __restrict__ b2, const float* __restrict__ b3,
                               const float* __restrict__ noise, float* __restrict__ z) {
    size_t i = (size_t)blockIdx.x * blockDim.x + threadIdx.x;
    if (i < (size_t)N_NODES * 64) {
        int row = (int)(i >> 6);
        int col = (int)(i & 63);
        float ci = c_in[row];
        float mean    = agg2[i] * ci + b2[col];
        float log_std = agg3[i] * ci + b3[col];
        z[i] = mean + noise[i] * __expf(log_std);
    }
}

// logits[g,c] = sum_j z[g*16384+j] * Wc[c*16384+j] + bc[c]
__global__ __launch_bounds__(256) void classifier_kernel(
        const float* __restrict__ z, const float* __restrict__ Wc,
        const float* __restrict__ bc, float* __restrict__ out) {
    __shared__ float s0[256];
    __shared__ float s1[256];
    const int g = blockIdx.x;
    const float* e = z + (size_t)g * 16384;
    float a0 = 0.0f, a1 = 0.0f;
    for (int j = threadIdx.x; j < 16384; j += 256) {
        float v = e[j];
        a0 += v * Wc[j];
        a1 += v * Wc[16384 + j];
    }
    s0[threadIdx.x] = a0;
    s1[threadIdx.x] = a1;
    __syncthreads();
    for (int st = 128; st > 0; st >>= 1) {
        if (threadIdx.x < st) {
            s0[threadIdx.x] += s0[threadIdx.x + st];
            s1[threadIdx.x] += s1[threadIdx.x + st];
        }
        __syncthreads();
    }
    if (threadIdx.x == 0) {
        out[g * 2 + 0] = s0[0] + bc[0];
        out[g * 2 + 1] = s1[0] + bc[1];
    }
}

// --------------------------------------------------------------------- driver

extern "C" void kernel_launch(void* const* d_in, const int* in_sizes, int n_in,
                              void* d_out, int out_size, void* d_ws, size_t ws_size,
                              hipStream_t stream) {
    const float* features = (const float*)d_in[0];   // [65536,256]
    const int*   src      = (const int*)  d_in[1];   // [2M]
    const int*   dst      = (const int*)  d_in[2];   // [2M]
    const float* W1       = (const float*)d_in[3];   // [256,128]
    const float* b1       = (const float*)d_in[4];   // [128]
    const float* W2       = (const float*)d_in[5];   // [128,64]
    const float* b2       = (const float*)d_in[6];   // [64]
    const float* W3       = (const float*)d_in[7];   // [128,64]
    const float* b3       = (const float*)d_in[8];   // [64]
    const float* Wc       = (const float*)d_in[9];   // [2,16384]
    const float* bc       = (const float*)d_in[10];  // [2]
    const float* noise    = (const float*)d_in[11];  // [65536,64]

    float* ws    = (float*)d_ws;
    float* c_out = ws;                                   // 65536
    float* c_in  = c_out + N_NODES;                      // 65536
    float* bufA  = c_in + N_NODES;                       // 65536*128  (gemm out)
    float* bufB  = bufA + (size_t)N_NODES * 128;         // 65536*128  (agg / h1)
    const size_t N64 = (size_t)N_NODES * 64;

    // 1) degree norms: c = rsqrt(max(deg,1))
    zero_f32<<<512, 256, 0, stream>>>(c_out, (size_t)2 * N_NODES);
    degree_kernel<<<N_EDGES / 256, 256, 0, stream>>>(src, dst, c_out, c_in, N_EDGES);
    invsqrt_kernel<<<(2 * N_NODES) / 256, 256, 0, stream>>>(c_out, 2 * N_NODES);

    // 2) layer 1: tmp = (x@W1)*c_out  -> aggregate over edges -> relu(+b1)
    gemm_f32_wmma<256, 8><<<N_NODES / 128, 256, 0, stream>>>(features, W1, c_out, bufA);
    zero_f32<<<2048, 256, 0, stream>>>(bufB, (size_t)N_NODES * 128);
    scatter_add<32><<<(int)(((size_t)N_EDGES * 32) / 256), 256, 0, stream>>>(bufA, src, dst, bufB, N_EDGES);
    bias_relu_inplace<128><<<(int)(((size_t)N_NODES * 128) / 256), 256, 0, stream>>>(
        bufB, c_in, b1, (size_t)N_NODES * 128);

    // 3) layers 2 & 3: mean/log_std pre-aggregation GEMMs (h1 = bufB)
    gemm_f32_wmma<128, 4><<<N_NODES / 128, 256, 0, stream>>>(bufB, W2, c_out, bufA);
    gemm_f32_wmma<128, 4><<<N_NODES / 128, 256, 0, stream>>>(bufB, W3, c_out, bufA + N64);

    // 4) aggregate both halves; bufB reused as agg2|agg3 (h1 no longer needed)
    zero_f32<<<2048, 256, 0, stream>>>(bufB, (size_t)N_NODES * 128);
    scatter_add<16><<<(int)(((size_t)N_EDGES * 16) / 256), 256, 0, stream>>>(bufA, src, dst, bufB, N_EDGES);
    scatter_add<16><<<(int)(((size_t)N_EDGES * 16) / 256), 256, 0, stream>>>(bufA + N64, src, dst, bufB + N64, N_EDGES);

    // 5) reparameterization: z written into bufA (free after scatter)
    reparam_kernel<<<(int)(N64 / 256), 256, 0, stream>>>(bufB, bufB + N64, c_in, b2, b3, noise, bufA);

    // 6) dense classifier
    classifier_kernel<<<256, 256, 0, stream>>>(bufA, Wc, bc, (float*)d_out);
}